// Focal_loss_ignoreneg_max_pos_44040594653639
// MI455X (gfx1250) — compile-verified
//
#include <hip/hip_runtime.h>

typedef float v2f __attribute__((ext_vector_type(2)));
typedef float v8f __attribute__((ext_vector_type(8)));

// ---------------------------------------------------------------------------
// Wave-level (32-lane) sum using V_WMMA_F32_16X16X4_F32 (full-f32 matrix op).
// A holds each lane's value in one 16x4 slot (other slot zero), B = ones.
// D(m,n) = rowsum_m for every n; C/D layout puts rows 0..7 in lanes 0..15 and
// rows 8..15 in lanes 16..31, so (sum of 8 D regs) + shfl_xor(16) = full sum
// in every lane. Exact f32 FMA order is fixed by HW -> deterministic.
// ---------------------------------------------------------------------------
__device__ __forceinline__ float wave_sum32(float v) {
#if __has_builtin(__builtin_amdgcn_wmma_f32_16x16x4_f32)
  v2f a; a[0] = v;    a[1] = 0.0f;
  v2f b; b[0] = 1.0f; b[1] = 1.0f;
  v8f c = {0.f, 0.f, 0.f, 0.f, 0.f, 0.f, 0.f, 0.f};
  v8f d = __builtin_amdgcn_wmma_f32_16x16x4_f32(
      /*neg_a=*/false, a, /*neg_b=*/false, b,
      /*c_mod=*/(short)0, c, /*reuse_a=*/false, /*reuse_b=*/false);
  float s = ((d[0] + d[1]) + (d[2] + d[3])) + ((d[4] + d[5]) + (d[6] + d[7]));
  s += __shfl_xor(s, 16, 32);
  return s;
#else
  #pragma unroll
  for (int off = 16; off > 0; off >>= 1) v += __shfl_xor(v, off, 32);
  return v;
#endif
}

// Block-wide reduction of (loss, count); thread 0 writes the 2-float partial.
// Must be called by ALL threads of the block (WMMA needs EXEC all-ones).
__device__ __forceinline__ void block_reduce2(float l, float c,
                                              float* __restrict__ out2) {
  __shared__ float sl[8];
  __shared__ float sc[8];
  l = wave_sum32(l);
  c = wave_sum32(c);
  const int wid  = threadIdx.x >> 5;
  const int lane = threadIdx.x & 31;
  if (lane == 0) { sl[wid] = l; sc[wid] = c; }
  __syncthreads();
  if (threadIdx.x == 0) {
    const int nw = (int)(blockDim.x >> 5);
    float L = 0.f, C = 0.f;
    for (int i = 0; i < nw; ++i) { L += sl[i]; C += sc[i]; }
    out2[0] = L;
    out2[1] = C;
  }
}

// Fast ~1ulp reciprocal (v_rcp_f32) with safe fallback.
__device__ __forceinline__ float fast_rcp(float x) {
#if __has_builtin(__builtin_amdgcn_rcpf)
  return __builtin_amdgcn_rcpf(x);
#else
  return 1.0f / x;
#endif
}

// ---------------------------------------------------------------------------
// Negative part: w = sigmoid(x)^2 * (prob==-1); loss += softplus(x)*w*factor
// Native-transcendental forms (v_exp_f32 / v_log_f32 / v_rcp_f32):
//   e      = exp(-|x|)            in (0, 1]  -> no over/underflow
//   sigmoid= x>=0 ? 1/(1+e) : e/(1+e)        (stable)
//   softplus= max(x,0) + log(1+e)            (1+e in (1,2] -> no cancellation)
// Cuts VALU work ~3x so the loop stays HBM-bound.
// ---------------------------------------------------------------------------
__device__ __forceinline__ void neg_elem(float x, float p, float& ll, float& lc) {
  const float m   = (p == -1.0f) ? 1.0f : 0.0f;
  const float e   = __expf(-fabsf(x));        // v_exp_f32 path
  const float l1p = __logf(1.0f + e);         // v_log_f32 path
  const float inv = fast_rcp(1.0f + e);       // v_rcp_f32
  const float sig = (x >= 0.0f) ? inv : e * inv;
  const float nll = fmaxf(x, 0.0f) + l1p;     // softplus(x) = -log_sigmoid(-x)
  const float w   = sig * sig * m;
  ll = fmaf(nll, w, ll);
  lc += w;
}

__device__ __forceinline__ void neg_f4(float4 x, float4 p, float& ll, float& lc) {
  neg_elem(x.x, p.x, ll, lc);
  neg_elem(x.y, p.y, ll, lc);
  neg_elem(x.z, p.z, ll, lc);
  neg_elem(x.w, p.w, ll, lc);
}

__global__ void neg_kernel(const float4* __restrict__ lg0, const float4* __restrict__ pr0,
                           const float4* __restrict__ lg1, const float4* __restrict__ pr1,
                           const float4* __restrict__ lg2, const float4* __restrict__ pr2,
                           float* __restrict__ partials) {
  const int tid    = blockIdx.x * blockDim.x + threadIdx.x;
  const int stride = gridDim.x * blockDim.x;

  float loss = 0.f, cnt = 0.f;

  {   // level 0, factor 4
    float ll = 0.f, lc = 0.f;
    const int n4 = (2 * 2 * 128 * 128 * 128) / 4;
    for (int i = tid; i < n4; i += stride) neg_f4(lg0[i], pr0[i], ll, lc);
    loss += ll * 4.0f; cnt += lc;
  }
  {   // level 1, factor 2
    float ll = 0.f, lc = 0.f;
    const int n4 = (2 * 2 * 64 * 64 * 64) / 4;
    for (int i = tid; i < n4; i += stride) neg_f4(lg1[i], pr1[i], ll, lc);
    loss += ll * 2.0f; cnt += lc;
  }
  {   // level 2, factor 1
    float ll = 0.f, lc = 0.f;
    const int n4 = (2 * 2 * 32 * 32 * 32) / 4;
    for (int i = tid; i < n4; i += stride) neg_f4(lg2[i], pr2[i], ll, lc);
    loss += ll; cnt += lc;
  }

  block_reduce2(loss, cnt, &partials[2 * blockIdx.x]);
}

// ---------------------------------------------------------------------------
// Positive part: one block per (level, batch), one thread per coord (P=128).
// Tiny (768 threads total) -> keep the accurate libm transcendentals here.
// ---------------------------------------------------------------------------
__global__ void pos_kernel(const float* __restrict__ lg0,
                           const float* __restrict__ lg1,
                           const float* __restrict__ lg2,
                           const int* __restrict__ co0,
                           const int* __restrict__ co1,
                           const int* __restrict__ co2,
                           float* __restrict__ partials) {
  const int lvl = (int)(blockIdx.x >> 1);
  const int b   = (int)(blockIdx.x & 1);

  const float* lg = (lvl == 0) ? lg0 : ((lvl == 1) ? lg1 : lg2);
  const int*   co = (lvl == 0) ? co0 : ((lvl == 1) ? co1 : co2);
  const int    D   = 128 >> lvl;                 // D = H = W
  const int    r   = lvl + 1;
  const float  fac = (float)(4 >> lvl);          // 4, 2, 1

  float loss = 0.f, cnt = 0.f;

  const int p = (int)threadIdx.x;                // 0..127
  const int* c = co + ((size_t)b * 128 + p) * 4;
  const int a = c[0], y = c[1], x = c[2], z = c[3];

  if (a > -1) {
    const int ac = (a < 0) ? 0 : ((a > 1) ? 1 : a);
    const int yc = (y < 0) ? 0 : ((y > D - 1) ? D - 1 : y);
    const int xc = (x < 0) ? 0 : ((x > D - 1) ? D - 1 : x);
    const int zc = (z < 0) ? 0 : ((z > D - 1) ? D - 1 : z);

    const float* base = lg + (size_t)(b * 2 + ac) * (size_t)D * D * D;

    float wmax = -__builtin_inff();
    for (int dy = -r; dy < r; ++dy) {
      int iy = yc + dy; iy = (iy < 0) ? 0 : ((iy > D - 1) ? D - 1 : iy);
      for (int dx = -r; dx < r; ++dx) {
        int ix = xc + dx; ix = (ix < 0) ? 0 : ((ix > D - 1) ? D - 1 : ix);
        const float* row = base + ((size_t)iy * D + ix) * (size_t)D;
        for (int dz = -r; dz < r; ++dz) {
          int iz = zc + dz; iz = (iz < 0) ? 0 : ((iz > D - 1) ? D - 1 : iz);
          wmax = fmaxf(wmax, row[iz]);
        }
      }
    }

    const float point    = base[((size_t)yc * D + xc) * (size_t)D + zc];
    const bool  near_low = (yc < r) || (xc < r) || (zc < r);
    const float lp       = near_low ? point : wmax;

    const float e      = expf(-fabsf(lp));
    const float l1p    = log1pf(e);
    const float logsig = fminf(lp, 0.0f) - l1p;          // log_sigmoid(lp)
    const float inv    = 1.0f / (1.0f + e);
    const float sig    = (lp >= 0.0f) ? inv : e * inv;   // sigmoid(lp)
    const float om     = 1.0f - sig;                     // match reference's literal 1-sigmoid
    const float w      = om * om;                        // ALPHA = 2
    loss = -logsig * w * fac;
    cnt  = w;
  }

  block_reduce2(loss, cnt, &partials[2 * blockIdx.x]);
}

// ---------------------------------------------------------------------------
// Final deterministic combine (double accumulation, fixed order).
// Output order: [pos_loss, neg_loss, pos_count, neg_count]
// ---------------------------------------------------------------------------
__global__ void final_kernel(const float* __restrict__ negPart, int nNegBlocks,
                             const float* __restrict__ posPart,
                             float* __restrict__ out) {
  __shared__ double sL[256];
  __shared__ double sC[256];
  const int t = (int)threadIdx.x;
  double l = 0.0, c = 0.0;
  for (int i = t; i < nNegBlocks; i += 256) {
    l += (double)negPart[2 * i];
    c += (double)negPart[2 * i + 1];
  }
  sL[t] = l; sC[t] = c;
  __syncthreads();
  if (t == 0) {
    double L = 0.0, C = 0.0;
    for (int i = 0; i < 256; ++i) { L += sL[i]; C += sC[i]; }
    double Lp = 0.0, Cp = 0.0;
    for (int i = 0; i < 6; ++i) { Lp += (double)posPart[2 * i]; Cp += (double)posPart[2 * i + 1]; }
    out[0] = (float)Lp;   // pos loss
    out[1] = (float)L;    // neg loss
    out[2] = (float)Cp;   // pos count
    out[3] = (float)C;    // neg count
  }
}

extern "C" void kernel_launch(void* const* d_in, const int* in_sizes, int n_in,
                              void* d_out, int out_size, void* d_ws, size_t ws_size,
                              hipStream_t stream) {
  (void)in_sizes; (void)n_in; (void)out_size;

  const float* lg0 = (const float*)d_in[0];
  const float* lg1 = (const float*)d_in[1];
  const float* lg2 = (const float*)d_in[2];
  const float* pr0 = (const float*)d_in[3];
  const float* pr1 = (const float*)d_in[4];
  const float* pr2 = (const float*)d_in[5];
  const int*   co0 = (const int*)d_in[6];
  const int*   co1 = (const int*)d_in[7];
  const int*   co2 = (const int*)d_in[8];

  // Workspace layout (floats): [0, 2*nblk) neg partials, then 12 pos partials.
  int nblk = 2048;
  const size_t avail = ws_size / sizeof(float);
  while (nblk > 32 && (size_t)(2 * nblk + 12) > avail) nblk >>= 1;

  float* negPart = (float*)d_ws;
  float* posPart = negPart + 2 * (size_t)nblk;

  neg_kernel<<<nblk, 256, 0, stream>>>(
      (const float4*)lg0, (const float4*)pr0,
      (const float4*)lg1, (const float4*)pr1,
      (const float4*)lg2, (const float4*)pr2, negPart);

  pos_kernel<<<6, 128, 0, stream>>>(lg0, lg1, lg2, co0, co1, co2, posPart);

  final_kernel<<<1, 256, 0, stream>>>(negPart, nblk, posPart, (float*)d_out);
}